// Poisson_learning_61220463837889
// MI455X (gfx1250) — compile-verified
//
#include <hip/hip_runtime.h>
#include <hip/hip_bf16.h>

// ---------------------------------------------------------------------------
// Poisson-learning pipeline for MI455X (gfx1250, wave32, WMMA).
//
// Roofline: ~75 GFLOP total vs ~300 MB HBM traffic -> memory-bound at
// 23.3 TB/s (~12 us). Therefore we keep full fp32 precision and use
// V_WMMA_F32_16X16X4_F32 for every matmul (softmax input precision matters,
// and fp32 WMMA is nowhere near the bottleneck here). A[8,1024,1024] (32 MB)
// fits in the 192 MB L2, so the 6 power-iteration passes re-read it from L2.
// ---------------------------------------------------------------------------

#define B_  8
#define L_  1024
#define H_  1024
#define K3_ 3072
#define N_  64

typedef __attribute__((ext_vector_type(2))) float v2f;
typedef __attribute__((ext_vector_type(8))) float v8f;

union Acc8 { v8f v; float f[8]; };

__device__ __forceinline__ v2f load2(const float* p) {
    return *(const v2f*)p;   // 8B-aligned by construction (even-k offsets)
}

__device__ __forceinline__ v8f wmma4(v2f a, v2f b, v8f c) {
    // D = A(16x4 f32) x B(4x16 f32) + C(16x16 f32)
    // 8 args: (neg_a, A, neg_b, B, c_mod, C, reuse_a, reuse_b)
    return __builtin_amdgcn_wmma_f32_16x16x4_f32(
        false, a, false, b, (short)0, c, false, false);
}

// ---------------------------------------------------------------------------
// K1: s[m,h] = sum_k samples[m,k] * W[h,k] + bias[h]
//     M = B*L = 8192, N = H = 1024, K = 3072. Both operands row-major along K.
// WG = 256 threads = 8 waves; WG tile 128(M) x 64(N); wave owns 16 x 64.
// ---------------------------------------------------------------------------
__global__ __launch_bounds__(256) void proj_kernel(
    const float* __restrict__ samples,   // [8192, 3072]
    const float* __restrict__ W,         // [1024, 3072]
    const float* __restrict__ bproj,     // [1024]
    float* __restrict__ s)               // [8192, 1024]
{
    const int lane = threadIdx.x & 31;
    const int wave = threadIdx.x >> 5;
    const int m0   = blockIdx.x * 128 + wave * 16;
    const int n0   = blockIdx.y * 64;
    const int kb   = (lane < 16) ? 0 : 2;     // A/B frag K base for this half
    const int rc   = lane & 15;               // row (A) / col (B) within tile

    const float* arow = samples + (size_t)(m0 + rc) * K3_;
    const float* br0  = W + (size_t)(n0 +  0 + rc) * K3_;
    const float* br1  = W + (size_t)(n0 + 16 + rc) * K3_;
    const float* br2  = W + (size_t)(n0 + 32 + rc) * K3_;
    const float* br3  = W + (size_t)(n0 + 48 + rc) * K3_;

    Acc8 c0, c1, c2, c3;
    #pragma unroll
    for (int r = 0; r < 8; ++r) { c0.f[r]=0.f; c1.f[r]=0.f; c2.f[r]=0.f; c3.f[r]=0.f; }

    for (int k = 0; k < K3_; k += 4) {
        v2f a  = load2(arow + k + kb);
        v2f b0 = load2(br0  + k + kb);
        v2f b1 = load2(br1  + k + kb);
        v2f b2 = load2(br2  + k + kb);
        v2f b3 = load2(br3  + k + kb);
        c0.v = wmma4(a, b0, c0.v);
        c1.v = wmma4(a, b1, c1.v);
        c2.v = wmma4(a, b2, c2.v);
        c3.v = wmma4(a, b3, c3.v);
    }

    const float bias0 = bproj[n0 +  0 + rc];
    const float bias1 = bproj[n0 + 16 + rc];
    const float bias2 = bproj[n0 + 32 + rc];
    const float bias3 = bproj[n0 + 48 + rc];

    #pragma unroll
    for (int r = 0; r < 8; ++r) {
        const int row = m0 + ((lane < 16) ? r : r + 8);
        float* out = s + (size_t)row * H_ + n0 + rc;
        out[ 0] = c0.f[r] + bias0;
        out[16] = c1.f[r] + bias1;
        out[32] = c2.f[r] + bias2;
        out[48] = c3.f[r] + bias3;
    }
}

// ---------------------------------------------------------------------------
// K2: fused graph + masked row-softmax + d.
//   G[l,m] = sum_h s[l,h]*s[m,h];  A = softmax(G - 1e5*I, axis=-1)
//   d[l]   = 1 / (sum_m A[l,m] + 1e-10)
// One WG per (batch, 16-row strip). 8 waves x 128 cols, 8 accumulators each.
// Softmax via width-16 shfl reductions + small LDS cross-wave reduce.
// ---------------------------------------------------------------------------
__global__ __launch_bounds__(256) void graph_softmax_kernel(
    const float* __restrict__ s,   // [B, L, H]
    float* __restrict__ A,         // [B, L, L]
    float* __restrict__ dvec)      // [B, L]
{
    const int b    = blockIdx.y;
    const int l0   = blockIdx.x * 16;
    const int lane = threadIdx.x & 31;
    const int wave = threadIdx.x >> 5;
    const int c0   = wave * 128;
    const int kb   = (lane < 16) ? 0 : 2;
    const int rc   = lane & 15;

    const float* sb   = s + (size_t)b * L_ * H_;
    const float* arow = sb + (size_t)(l0 + rc) * H_;

    const float* brow[8];
    #pragma unroll
    for (int ct = 0; ct < 8; ++ct)
        brow[ct] = sb + (size_t)(c0 + ct * 16 + rc) * H_;

    Acc8 acc[8];
    #pragma unroll
    for (int ct = 0; ct < 8; ++ct)
        #pragma unroll
        for (int r = 0; r < 8; ++r) acc[ct].f[r] = 0.f;

    for (int k = 0; k < H_; k += 4) {
        v2f a = load2(arow + k + kb);
        #pragma unroll
        for (int ct = 0; ct < 8; ++ct) {
            v2f bb = load2(brow[ct] + k + kb);
            acc[ct].v = wmma4(a, bb, acc[ct].v);
        }
    }

    // diagonal mask: element acc[ct].f[r] is (row = l0 + r or r+8, col)
    #pragma unroll
    for (int ct = 0; ct < 8; ++ct) {
        const int col = c0 + ct * 16 + rc;
        #pragma unroll
        for (int r = 0; r < 8; ++r) {
            const int row = l0 + ((lane < 16) ? r : r + 8);
            if (col == row) acc[ct].f[r] -= 1e5f;
        }
    }

    __shared__ float red[8][16];
    __shared__ float stat[16];

    // ---- pass 1: row max ----
    {
        float m[8];
        #pragma unroll
        for (int r = 0; r < 8; ++r) {
            float v = acc[0].f[r];
            #pragma unroll
            for (int ct = 1; ct < 8; ++ct) v = fmaxf(v, acc[ct].f[r]);
            #pragma unroll
            for (int off = 8; off >= 1; off >>= 1)
                v = fmaxf(v, __shfl_xor(v, off, 16));
            m[r] = v;    // every lane in its 16-lane half now holds the half-row max
        }
        if ((lane & 15) == 0) {
            const int half = lane >> 4;   // 0 -> rows r, 1 -> rows r+8
            #pragma unroll
            for (int r = 0; r < 8; ++r) red[wave][r + half * 8] = m[r];
        }
    }
    __syncthreads();
    if (threadIdx.x < 16) {
        float v = red[0][threadIdx.x];
        #pragma unroll
        for (int w = 1; w < 8; ++w) v = fmaxf(v, red[w][threadIdx.x]);
        stat[threadIdx.x] = v;
    }
    __syncthreads();

    // ---- pass 2: exponentiate + row sum ----
    {
        float ssum[8];
        #pragma unroll
        for (int r = 0; r < 8; ++r) {
            const int rl = (lane < 16) ? r : r + 8;
            const float mx = stat[rl];
            float t = 0.f;
            #pragma unroll
            for (int ct = 0; ct < 8; ++ct) {
                float e = __expf(acc[ct].f[r] - mx);
                acc[ct].f[r] = e;
                t += e;
            }
            #pragma unroll
            for (int off = 8; off >= 1; off >>= 1)
                t += __shfl_xor(t, off, 16);
            ssum[r] = t;
        }
        if ((lane & 15) == 0) {
            const int half = lane >> 4;
            #pragma unroll
            for (int r = 0; r < 8; ++r) red[wave][r + half * 8] = ssum[r];
        }
    }
    __syncthreads();
    if (threadIdx.x < 16) {
        float v = 0.f;
        #pragma unroll
        for (int w = 0; w < 8; ++w) v += red[w][threadIdx.x];
        stat[threadIdx.x] = v;
    }
    __syncthreads();

    // ---- pass 3: normalize, write A, accumulate actual row sum for d ----
    {
        float asum[8];
        float* Ab = A + (size_t)b * L_ * L_;
        #pragma unroll
        for (int r = 0; r < 8; ++r) {
            const int rl  = (lane < 16) ? r : r + 8;
            const int row = l0 + rl;
            const float inv = 1.0f / stat[rl];
            float t = 0.f;
            #pragma unroll
            for (int ct = 0; ct < 8; ++ct) {
                const int col = c0 + ct * 16 + rc;
                const float a = acc[ct].f[r] * inv;
                Ab[(size_t)row * L_ + col] = a;
                t += a;
            }
            #pragma unroll
            for (int off = 8; off >= 1; off >>= 1)
                t += __shfl_xor(t, off, 16);
            asum[r] = t;
        }
        if ((lane & 15) == 0) {
            const int half = lane >> 4;
            #pragma unroll
            for (int r = 0; r < 8; ++r) red[wave][r + half * 8] = asum[r];
        }
    }
    __syncthreads();
    if (threadIdx.x < 16) {
        float v = 0.f;
        #pragma unroll
        for (int w = 0; w < 8; ++w) v += red[w][threadIdx.x];
        dvec[b * L_ + l0 + threadIdx.x] = 1.0f / (v + 1e-10f);
    }
}

// ---------------------------------------------------------------------------
// K3: Db = d * (sup - avg) * row_has   (label centering; cheap VALU kernel)
// ---------------------------------------------------------------------------
__global__ __launch_bounds__(256) void center_kernel(
    const float* __restrict__ sup,   // [B, L, N]
    const float* __restrict__ dvec,  // [B, L]
    float* __restrict__ Db)          // [B, L, N]
{
    const int b   = blockIdx.x;
    const int tid = threadIdx.x;
    __shared__ float colsum[4][64];
    __shared__ float rowhas[L_];
    __shared__ float avg[64];
    __shared__ float totalsh;

    const float* S = sup + (size_t)b * L_ * N_;
    if (tid == 0) totalsh = 0.f;
    __syncthreads();

    // column partial sums (coalesced)
    {
        const int n = tid & 63, g = tid >> 6;
        float cs = 0.f;
        for (int l = g; l < L_; l += 4) cs += S[l * N_ + n];
        colsum[g][n] = cs;
    }
    // row flags + total
    {
        float tpart = 0.f;
        for (int l = tid; l < L_; l += 256) {
            float rs = 0.f;
            #pragma unroll 8
            for (int n = 0; n < N_; ++n) rs += S[l * N_ + n];
            const float flag = (rs > 0.5f) ? 1.f : 0.f;
            rowhas[l] = flag;
            tpart += flag;
        }
        atomicAdd(&totalsh, tpart);
    }
    __syncthreads();
    if (tid < 64)
        avg[tid] = (colsum[0][tid] + colsum[1][tid] + colsum[2][tid] + colsum[3][tid]) / totalsh;
    __syncthreads();

    const float* db = dvec + b * L_;
    float* O = Db + (size_t)b * L_ * N_;
    for (int e = tid; e < L_ * N_; e += 256) {
        const int l = e >> 6, n = e & 63;
        O[e] = db[l] * (S[e] - avg[n]) * rowhas[l];
    }
}

// ---------------------------------------------------------------------------
// K4: one power-iteration step:
//   ut_out[l,n] = d[l] * sum_m A[m,l] * ut_in[m,n] + Db[l,n] (+ predict)
// A read transposed (column-strided fragment loads). One wave per 16x64 tile.
// ---------------------------------------------------------------------------
__global__ __launch_bounds__(256) void iter_kernel(
    const float* __restrict__ Amat,    // [B, L, L]
    const float* __restrict__ dvec,    // [B, L]
    const float* __restrict__ Db,      // [B, L, N]
    const float* __restrict__ ut_in,   // [B, L, N]
    const float* __restrict__ predict, // [B, L, N]
    float* __restrict__ ut_out,        // [B, L, N]
    int add_predict)
{
    const int gw   = (int)((blockIdx.x * 256 + threadIdx.x) >> 5); // 0..511
    const int b    = gw >> 6;
    const int l0   = (gw & 63) * 16;
    const int lane = threadIdx.x & 31;
    const int kb   = (lane < 16) ? 0 : 2;
    const int rc   = lane & 15;

    const float* Ab = Amat + (size_t)b * L_ * L_;
    const float* U  = ut_in + (size_t)b * L_ * N_;

    Acc8 acc[4];
    #pragma unroll
    for (int ct = 0; ct < 4; ++ct)
        #pragma unroll
        for (int r = 0; r < 8; ++r) acc[ct].f[r] = 0.f;

    for (int k = 0; k < L_; k += 4) {
        // A-frag element (row=l0+rc, kk) = Amat[k+kb+kk][l0+rc] (transposed read)
        const float* ap = Ab + (size_t)(k + kb) * L_ + (l0 + rc);
        v2f a = { ap[0], ap[L_] };
        const float* up = U + (size_t)(k + kb) * N_;
        #pragma unroll
        for (int ct = 0; ct < 4; ++ct) {
            v2f bb = { up[ct * 16 + rc], up[N_ + ct * 16 + rc] };
            acc[ct].v = wmma4(a, bb, acc[ct].v);
        }
    }

    const float* Dbb = Db + (size_t)b * L_ * N_;
    const float* Pb  = predict + (size_t)b * L_ * N_;
    float* Ob = ut_out + (size_t)b * L_ * N_;

    #pragma unroll
    for (int r = 0; r < 8; ++r) {
        const int row = l0 + ((lane < 16) ? r : r + 8);
        const float dl = dvec[b * L_ + row];
        #pragma unroll
        for (int ct = 0; ct < 4; ++ct) {
            const int col = ct * 16 + rc;
            float y = dl * acc[ct].f[r] + Dbb[(size_t)row * N_ + col];
            if (add_predict) y += Pb[(size_t)row * N_ + col];
            Ob[(size_t)row * N_ + col] = y;
        }
    }
}

__global__ void zero_kernel(float* p, int n) {
    const int i = blockIdx.x * 256 + threadIdx.x;
    if (i < n) p[i] = 0.f;
}

// ---------------------------------------------------------------------------
extern "C" void kernel_launch(void* const* d_in, const int* in_sizes, int n_in,
                              void* d_out, int out_size, void* d_ws, size_t ws_size,
                              hipStream_t stream) {
    const float* samples = (const float*)d_in[0]; // [8,1024,3072]
    const float* labels  = (const float*)d_in[1]; // [8,1024,64]
    const float* predict = (const float*)d_in[2]; // [8,1024,64]
    const float* W       = (const float*)d_in[3]; // [1024,3072]
    const float* bproj   = (const float*)d_in[4]; // [1024]

    float* ws = (float*)d_ws;
    // workspace layout (floats):
    float* s    = ws;                               // 8192*1024      = 8388608
    float* Amat = s    + (size_t)8192 * 1024;       // 8*1024*1024    = 8388608
    float* dvec = Amat + (size_t)8 * 1024 * 1024;   // 8*1024         = 8192
    float* Db   = dvec + (size_t)8 * 1024;          // 8*1024*64      = 524288
    float* ut0  = Db   + (size_t)8 * 1024 * 64;     // 524288
    float* ut1  = ut0  + (size_t)8 * 1024 * 64;     // 524288
    // total ~73.5 MB of d_ws

    // K1: projection GEMM (WMMA f32)
    proj_kernel<<<dim3(8192 / 128, 1024 / 64), 256, 0, stream>>>(samples, W, bproj, s);

    // K2: fused graph + softmax + d (WMMA f32)
    graph_softmax_kernel<<<dim3(L_ / 16, B_), 256, 0, stream>>>(s, Amat, dvec);

    // K3: label centering
    center_kernel<<<dim3(B_), 256, 0, stream>>>(labels, dvec, Db);

    // zero ut0 (initial state)
    zero_kernel<<<dim3((8 * 1024 * 64 + 255) / 256), 256, 0, stream>>>(ut0, 8 * 1024 * 64);

    // K4: six power-iteration steps (WMMA f32), predict injected at t==3,
    // last step writes straight to d_out.
    const float* cur = ut0;
    for (int t = 1; t <= 6; ++t) {
        float* out = (t == 6) ? (float*)d_out : ((t & 1) ? ut1 : ut0);
        iter_kernel<<<dim3(512 / 8), 256, 0, stream>>>(
            Amat, dvec, Db, cur, predict, out, (t == 3) ? 1 : 0);
        cur = out;
    }
}